// FeatureSubSelfAttention_67345087201655
// MI455X (gfx1250) — compile-verified
//
#include <hip/hip_runtime.h>
#include <hip/hip_bf16.h>
#include <math.h>
#include <stdint.h>

// FeatureSubSelfAttention for MI455X (gfx1250).
// wave32 + V_WMMA_F32_16X16X4_F32 + GLOBAL_LOAD_ASYNC_TO_LDS double buffering.
// B=16, C=256, H=W=64 -> N=4096, HEADS=4.

typedef __attribute__((ext_vector_type(2))) float v2f;
typedef __attribute__((ext_vector_type(8))) float v8f;

#define BATCH 16
#define CCH   256
#define NPIX  4096
#define OKVL  768

__device__ __forceinline__ float gelu_exact(float v) {
    return 0.5f * v * (1.0f + erff(v * 0.70710678118654752440f));
}

__device__ __forceinline__ v8f wmma_f32(v2f a, v2f b, v8f c) {
    // V_WMMA_F32_16X16X4_F32 : D = A(16x4) * B(4x16) + C(16x16)
    return __builtin_amdgcn_wmma_f32_16x16x4_f32(
        false, a, false, b, (short)0, c, false, false);
}

// Low 32 bits of a flat pointer into the LDS aperture == wave-relative LDS
// byte offset (ISA 10.2).
__device__ __forceinline__ uint32_t lds_off32(const void* p) {
    return (uint32_t)(uintptr_t)p;
}

// Async memory->LDS copy, 16B per lane, tracked by ASYNCcnt (no VGPR data).
__device__ __forceinline__ void async_b128(uint32_t lds, const float* g) {
    asm volatile("global_load_async_to_lds_b128 %0, %1, off"
                 :: "v"(lds), "v"(g) : "memory");
}
__device__ __forceinline__ void wait_async0() {
    asm volatile("s_wait_asynccnt 0x0" ::: "memory");
}

// ---------------------------------------------------------------------------
// Compute helpers: one 16-deep K chunk of 8 wmma tiles per wave.
// A operand: row-major (m x k) LDS tile, row stride 20 (16B aligned, no bank
// conflicts).  B operand either (k x n) [16][132] or row-major (n x k) [128][20].
// ---------------------------------------------------------------------------
__device__ __forceinline__ void mma_chunk_wx(
    const float (*Wt)[20], const float (*Xt)[132],
    int wo, int wn, int l15, int khalf, v8f (&acc)[4][2])
{
    #pragma unroll
    for (int kk = 0; kk < 16; kk += 4) {
        v2f a[4], bf[2];
        #pragma unroll
        for (int mt = 0; mt < 4; ++mt) {
            const int m = wo*64 + mt*16 + l15;
            a[mt].x = Wt[m][kk + khalf];
            a[mt].y = Wt[m][kk + khalf + 1];
        }
        #pragma unroll
        for (int nt = 0; nt < 2; ++nt) {
            const int n = wn*32 + nt*16 + l15;
            bf[nt].x = Xt[kk + khalf][n];
            bf[nt].y = Xt[kk + khalf + 1][n];
        }
        #pragma unroll
        for (int mt = 0; mt < 4; ++mt)
            #pragma unroll
            for (int nt = 0; nt < 2; ++nt)
                acc[mt][nt] = wmma_f32(a[mt], bf[nt], acc[mt][nt]);
    }
}

__device__ __forceinline__ void mma_chunk_nt(
    const float (*Qt)[20], const float (*Kt)[20],
    int wo, int wn, int l15, int khalf, v8f (&acc)[4][2])
{
    #pragma unroll
    for (int kk = 0; kk < 16; kk += 4) {
        v2f a[4], bf[2];
        #pragma unroll
        for (int mt = 0; mt < 4; ++mt) {
            const int m = wo*64 + mt*16 + l15;
            a[mt].x = Qt[m][kk + khalf];
            a[mt].y = Qt[m][kk + khalf + 1];
        }
        #pragma unroll
        for (int nt = 0; nt < 2; ++nt) {
            const int n = wn*32 + nt*16 + l15;
            bf[nt].x = Kt[n][kk + khalf];
            bf[nt].y = Kt[n][kk + khalf + 1];
        }
        #pragma unroll
        for (int mt = 0; mt < 4; ++mt)
            #pragma unroll
            for (int nt = 0; nt < 2; ++nt)
                acc[mt][nt] = wmma_f32(a[mt], bf[nt], acc[mt][nt]);
    }
}

// ---------------------------------------------------------------------------
// Y[b] = act(W @ X[b] + bias).  W:(O,Kdim) ld=ldw.  X:(B,Kdim,N).
// 128x128 block tile, 8 waves x (64x32), double-buffered async LDS pipeline.
// ---------------------------------------------------------------------------
template<bool GELU, bool HAS_BIAS>
__global__ __launch_bounds__(256)
void gemm_wx(const float* __restrict__ W, int ldw,
             const float* __restrict__ X, long long xstride,
             const float* __restrict__ bias,
             float* __restrict__ Y, long long ystride,
             int Kdim, int N)
{
    const int b  = blockIdx.z;
    const int n0 = blockIdx.x * 128;
    const int o0 = blockIdx.y * 128;
    const float* Xb = X + (long long)b * xstride;
    float*       Yb = Y + (long long)b * ystride;

    __shared__ float Wt[2][128][20];
    __shared__ float Xt[2][16][132];

    const int tid   = threadIdx.x;
    const int lane  = tid & 31;
    const int wave  = tid >> 5;
    const int wo    = wave >> 2;
    const int wn    = wave & 3;
    const int l15   = lane & 15;
    const int khalf = (lane >> 4) << 1;

    v8f acc[4][2];
    #pragma unroll
    for (int i = 0; i < 4; ++i)
        #pragma unroll
        for (int j = 0; j < 2; ++j)
            acc[i][j] = (v8f){0.f,0.f,0.f,0.f,0.f,0.f,0.f,0.f};

    const int wr = tid >> 1;           // 0..127
    const int wc = (tid & 1) << 3;     // 0 or 8
    const int xr = tid >> 4;           // 0..15
    const int xc = (tid & 15) << 3;    // 0..120

    auto load_chunk = [&](int buf, int k0) {
        const float* wp = W + (size_t)(o0 + wr) * ldw + k0 + wc;
        async_b128(lds_off32(&Wt[buf][wr][wc]),     wp);
        async_b128(lds_off32(&Wt[buf][wr][wc + 4]), wp + 4);
        const float* xp = Xb + (size_t)(k0 + xr) * N + n0 + xc;
        async_b128(lds_off32(&Xt[buf][xr][xc]),     xp);
        async_b128(lds_off32(&Xt[buf][xr][xc + 4]), xp + 4);
    };

    load_chunk(0, 0);
    wait_async0();
    __syncthreads();

    int cur = 0;
    for (int k0 = 0; k0 < Kdim; k0 += 16) {
        const int nxt = cur ^ 1;
        if (k0 + 16 < Kdim) load_chunk(nxt, k0 + 16);
        mma_chunk_wx(Wt[cur], Xt[cur], wo, wn, l15, khalf, acc);
        wait_async0();
        __syncthreads();
        cur = nxt;
    }

    #pragma unroll
    for (int mt = 0; mt < 4; ++mt) {
        const int mbase = o0 + wo*64 + mt*16 + ((lane >> 4) << 3);
        #pragma unroll
        for (int nt = 0; nt < 2; ++nt) {
            const int n = n0 + wn*32 + nt*16 + l15;
            #pragma unroll
            for (int r = 0; r < 8; ++r) {
                const int m = mbase + r;
                float v = acc[mt][nt][r];
                if (HAS_BIAS) v += bias[m];
                if (GELU)     v = gelu_exact(v);
                Yb[(size_t)m * N + n] = v;
            }
        }
    }
}

// ---------------------------------------------------------------------------
// QK[b][c][d] = sum_n Q[b][c][n] * K[b][d][n]  (Q @ K^T), reduction N=4096.
// ---------------------------------------------------------------------------
__global__ __launch_bounds__(256)
void gemm_qkT(const float* __restrict__ Q, long long qstride,
              const float* __restrict__ Kmat, long long kstride,
              float* __restrict__ QK)
{
    const int b  = blockIdx.z;
    const int d0 = blockIdx.x * 128;
    const int c0 = blockIdx.y * 128;
    const float* Qb = Q    + (long long)b * qstride;
    const float* Kb = Kmat + (long long)b * kstride;

    __shared__ float Qt[2][128][20];
    __shared__ float Kt[2][128][20];

    const int tid   = threadIdx.x;
    const int lane  = tid & 31;
    const int wave  = tid >> 5;
    const int wo    = wave >> 2;
    const int wn    = wave & 3;
    const int l15   = lane & 15;
    const int khalf = (lane >> 4) << 1;

    v8f acc[4][2];
    #pragma unroll
    for (int i = 0; i < 4; ++i)
        #pragma unroll
        for (int j = 0; j < 2; ++j)
            acc[i][j] = (v8f){0.f,0.f,0.f,0.f,0.f,0.f,0.f,0.f};

    const int lr = tid >> 1;           // 0..127
    const int lc = (tid & 1) << 3;     // 0 or 8

    auto load_chunk = [&](int buf, int nk) {
        const float* qp = Qb + (size_t)(c0 + lr) * NPIX + nk + lc;
        async_b128(lds_off32(&Qt[buf][lr][lc]),     qp);
        async_b128(lds_off32(&Qt[buf][lr][lc + 4]), qp + 4);
        const float* kp = Kb + (size_t)(d0 + lr) * NPIX + nk + lc;
        async_b128(lds_off32(&Kt[buf][lr][lc]),     kp);
        async_b128(lds_off32(&Kt[buf][lr][lc + 4]), kp + 4);
    };

    load_chunk(0, 0);
    wait_async0();
    __syncthreads();

    int cur = 0;
    for (int nk = 0; nk < NPIX; nk += 16) {
        const int nxt = cur ^ 1;
        if (nk + 16 < NPIX) load_chunk(nxt, nk + 16);
        mma_chunk_nt(Qt[cur], Kt[cur], wo, wn, l15, khalf, acc);
        wait_async0();
        __syncthreads();
        cur = nxt;
    }

    float* QKb = QK + (size_t)b * CCH * CCH;
    #pragma unroll
    for (int mt = 0; mt < 4; ++mt) {
        const int mbase = c0 + wo*64 + mt*16 + ((lane >> 4) << 3);
        #pragma unroll
        for (int nt = 0; nt < 2; ++nt) {
            const int n = d0 + wn*32 + nt*16 + l15;
            #pragma unroll
            for (int r = 0; r < 8; ++r)
                QKb[(size_t)(mbase + r) * CCH + n] = acc[mt][nt][r];
        }
    }
}

// ---------------------------------------------------------------------------
// Row softmax over 256 elements, one block per row (B*256 rows).
// ---------------------------------------------------------------------------
__global__ __launch_bounds__(256)
void softmax256(float* __restrict__ qk)
{
    const int row = blockIdx.x;
    float* p = qk + (size_t)row * 256;
    const int tid = threadIdx.x;
    __shared__ float red[256];

    float v = p[tid];
    red[tid] = v;
    __syncthreads();
    for (int s = 128; s > 0; s >>= 1) {
        if (tid < s) red[tid] = fmaxf(red[tid], red[tid + s]);
        __syncthreads();
    }
    const float m = red[0];
    __syncthreads();
    const float e = expf(v - m);
    red[tid] = e;
    __syncthreads();
    for (int s = 128; s > 0; s >>= 1) {
        if (tid < s) red[tid] += red[tid + s];
        __syncthreads();
    }
    p[tid] = e / red[0];
}

// ---------------------------------------------------------------------------
// out[b] = gelu( Wl @ lfeat[b] + M[b] @ V[b] + b_proj )
//   Wl = w_proj[:, :256] (ld 512), lfeat = kvl[b,0], V = kvl[b,2],
//   M[b] = Wx @ softmax(qk)[b].  Single K=512 GEMM with dual-source K loop.
// ---------------------------------------------------------------------------
__global__ __launch_bounds__(256)
void gemm_final(const float* __restrict__ Wproj,   // (256, 512)
                const float* __restrict__ Mws,     // (B, 256, 256)
                const float* __restrict__ kvl,     // (B, 768, 4096)
                const float* __restrict__ bias,    // (256)
                float* __restrict__ Y)             // (B, 256, 4096)
{
    const int b  = blockIdx.z;
    const int n0 = blockIdx.x * 128;
    const int o0 = blockIdx.y * 128;
    const float* lf = kvl + (size_t)b * OKVL * NPIX;     // lfeat plane
    const float* vv = lf  + (size_t)2 * CCH * NPIX;      // v plane
    const float* Mb = Mws + (size_t)b * CCH * CCH;
    float*       Yb = Y   + (size_t)b * CCH * NPIX;

    __shared__ float Wt[2][128][20];
    __shared__ float Xt[2][16][132];

    const int tid   = threadIdx.x;
    const int lane  = tid & 31;
    const int wave  = tid >> 5;
    const int wo    = wave >> 2;
    const int wn    = wave & 3;
    const int l15   = lane & 15;
    const int khalf = (lane >> 4) << 1;

    v8f acc[4][2];
    #pragma unroll
    for (int i = 0; i < 4; ++i)
        #pragma unroll
        for (int j = 0; j < 2; ++j)
            acc[i][j] = (v8f){0.f,0.f,0.f,0.f,0.f,0.f,0.f,0.f};

    const int wr = tid >> 1;
    const int wc = (tid & 1) << 3;
    const int xr = tid >> 4;
    const int xc = (tid & 15) << 3;

    auto load_chunk = [&](int buf, int k0) {
        const bool first = (k0 < CCH);
        const int  kq    = first ? k0 : (k0 - CCH);
        const float* wp = first ? (Wproj + (size_t)(o0 + wr) * 512 + kq + wc)
                                : (Mb    + (size_t)(o0 + wr) * 256 + kq + wc);
        async_b128(lds_off32(&Wt[buf][wr][wc]),     wp);
        async_b128(lds_off32(&Wt[buf][wr][wc + 4]), wp + 4);
        const float* xp = (first ? lf : vv) + (size_t)(kq + xr) * NPIX + n0 + xc;
        async_b128(lds_off32(&Xt[buf][xr][xc]),     xp);
        async_b128(lds_off32(&Xt[buf][xr][xc + 4]), xp + 4);
    };

    load_chunk(0, 0);
    wait_async0();
    __syncthreads();

    int cur = 0;
    for (int k0 = 0; k0 < 2 * CCH; k0 += 16) {
        const int nxt = cur ^ 1;
        if (k0 + 16 < 2 * CCH) load_chunk(nxt, k0 + 16);
        mma_chunk_wx(Wt[cur], Xt[cur], wo, wn, l15, khalf, acc);
        wait_async0();
        __syncthreads();
        cur = nxt;
    }

    #pragma unroll
    for (int mt = 0; mt < 4; ++mt) {
        const int mbase = o0 + wo*64 + mt*16 + ((lane >> 4) << 3);
        #pragma unroll
        for (int nt = 0; nt < 2; ++nt) {
            const int n = n0 + wn*32 + nt*16 + l15;
            #pragma unroll
            for (int r = 0; r < 8; ++r) {
                const int m = mbase + r;
                float v = acc[mt][nt][r] + bias[m];
                Yb[(size_t)m * NPIX + n] = gelu_exact(v);
            }
        }
    }
}

// ---------------------------------------------------------------------------
extern "C" void kernel_launch(void* const* d_in, const int* in_sizes, int n_in,
                              void* d_out, int out_size, void* d_ws, size_t ws_size,
                              hipStream_t stream) {
    const float* x      = (const float*)d_in[0];   // (B, C, H, W)
    const float* q_in   = (const float*)d_in[1];   // (B, C, H, W)
    const float* w_kvl  = (const float*)d_in[2];   // (768, 256)
    const float* b_kvl  = (const float*)d_in[3];   // (768)
    const float* w_q    = (const float*)d_in[4];   // (256, 256)
    const float* b_q    = (const float*)d_in[5];   // (256)
    const float* w_proj = (const float*)d_in[6];   // (256, 512)
    const float* b_proj = (const float*)d_in[7];   // (256)
    float* out = (float*)d_out;                    // (B, 256, 64, 64)

    float* ws  = (float*)d_ws;
    float* kvl = ws;                                       // B*768*4096
    float* q   = kvl + (size_t)BATCH * OKVL * NPIX;        // B*256*4096
    float* qk  = q   + (size_t)BATCH * CCH  * NPIX;        // B*256*256
    float* Mb  = qk  + (size_t)BATCH * CCH  * CCH;         // B*256*256

    dim3 blk(256);

    // 1) kvl = gelu(W_kvl @ x + b_kvl)       O=768, K=256, N=4096
    gemm_wx<true, true><<<dim3(NPIX/128, OKVL/128, BATCH), blk, 0, stream>>>(
        w_kvl, CCH, x, (long long)CCH * NPIX, b_kvl,
        kvl, (long long)OKVL * NPIX, CCH, NPIX);

    // 2) q = W_q @ q_input + b_q             O=256, K=256, N=4096
    gemm_wx<false, true><<<dim3(NPIX/128, CCH/128, BATCH), blk, 0, stream>>>(
        w_q, CCH, q_in, (long long)CCH * NPIX, b_q,
        q, (long long)CCH * NPIX, CCH, NPIX);

    // 3) qk = Q @ K^T                         256x256 per batch, K over N=4096
    gemm_qkT<<<dim3(CCH/128, CCH/128, BATCH), blk, 0, stream>>>(
        q, (long long)CCH * NPIX,
        kvl + (size_t)CCH * NPIX /* k plane of batch 0 */, (long long)OKVL * NPIX,
        qk);

    // 4) P = softmax(qk) rows (in place)
    softmax256<<<dim3(BATCH * CCH), blk, 0, stream>>>(qk);

    // 5) M = Wx @ P  (Wx = w_proj[:, 256:], ld 512)   O=256, K=256, N=256
    gemm_wx<false, false><<<dim3(CCH/128, CCH/128, BATCH), blk, 0, stream>>>(
        w_proj + CCH, 2 * CCH, qk, (long long)CCH * CCH, nullptr,
        Mb, (long long)CCH * CCH, CCH, CCH);

    // 6) out = gelu(Wl @ lfeat + M @ V + b_proj)
    gemm_final<<<dim3(NPIX/128, CCH/128, BATCH), blk, 0, stream>>>(
        w_proj, Mb, kvl, b_proj, out);
}